// simple_GAT_36893769073202
// MI455X (gfx1250) — compile-verified
//
#include <hip/hip_runtime.h>

typedef __attribute__((ext_vector_type(2))) float v2f;
typedef __attribute__((ext_vector_type(8))) float v8f;

#define NN 100000
#define EE 1600000
#define NEG_SLOPE 0.2f

// ---------- order-preserving float <-> uint for atomicMax-based segment max ----
__device__ __forceinline__ unsigned f2ord(float f) {
  unsigned u = __float_as_uint(f);
  return (u & 0x80000000u) ? ~u : (u | 0x80000000u);
}
__device__ __forceinline__ float ord2f(unsigned u) {
  return __uint_as_float((u & 0x80000000u) ? (u & 0x7FFFFFFFu) : ~u);
}

// ---------------- generic zero ----------------
__global__ void zero_f32(float* __restrict__ p, long n) {
  for (long i = (long)blockIdx.x * blockDim.x + threadIdx.x; i < n;
       i += (long)gridDim.x * blockDim.x)
    p[i] = 0.0f;
}

// ---------------- sum-reduce edge_attr (for self-loop fill_value='mean') ------
__global__ __launch_bounds__(256) void reduce_sum(const float* __restrict__ in,
                                                  float* __restrict__ out, int n) {
  float acc = 0.0f;
  for (long i = (long)blockIdx.x * blockDim.x + threadIdx.x; i < n;
       i += (long)gridDim.x * blockDim.x)
    acc += in[i];
  for (int o = 16; o > 0; o >>= 1) acc += __shfl_xor(acc, o, 32);
  __shared__ float sm[8];
  int lane = threadIdx.x & 31, w = threadIdx.x >> 5;
  if (lane == 0) sm[w] = acc;
  __syncthreads();
  if (threadIdx.x == 0) {
    float s = 0.0f;
    for (int i = 0; i < 8; ++i) s += sm[i];
    atomicAdd(out, s);
  }
}

// ---------------- WMMA f32 GEMM: Y[N,64] = X[N,FI] @ W[FI,64] + bias ----------
// One wave computes one 16-row x 64-col tile with 4x V_WMMA_F32_16X16X4_F32
// accumulators, K stepped by 4. Fragment layouts per CDNA5 ISA 7.12.2:
//   A 16x4 (2 VGPR): lanes0-15 row M, {K=0,K=1}; lanes16-31 row M, {K=2,K=3}
//   B 4x16 (2 VGPR): lanes0-15 col N, rows {kk, kk+1}; lanes16-31 rows {kk+2,+3}
//   C/D 16x16 (8 VGPR): vgpr v -> M = v + (lane>=16 ? 8 : 0), N = lane&15
template <int FI>
__global__ __launch_bounds__(128) void gemm_wmma_f32(
    const float* __restrict__ X, const float* __restrict__ W,
    const float* __restrict__ bias, float* __restrict__ Y, int n_rows) {
  const int lane = threadIdx.x & 31;
  const int wave = threadIdx.x >> 5;
  const int tile = blockIdx.x * 4 + wave;  // 16-row tile (wave-uniform)
  if (tile * 16 >= n_rows) return;         // uniform exit: EXEC all-1 at WMMA
  const int row0 = tile * 16;
  const int m  = lane & 15;           // row (A) / col (B,C)
  const int kk = (lane >> 4) << 1;    // 0 or 2: K sub-offset for this half-wave

  v8f acc[4] = {v8f{0}, v8f{0}, v8f{0}, v8f{0}};

#pragma unroll
  for (int k = 0; k < FI; k += 4) {
    v2f a;
    const float* xp = X + (size_t)(row0 + m) * FI + (k + kk);
    a.x = xp[0];
    a.y = xp[1];
#pragma unroll
    for (int c = 0; c < 4; ++c) {
      v2f b;
      const float* wp = W + (size_t)(k + kk) * 64 + c * 16 + m;
      b.x = wp[0];
      b.y = wp[64];
      acc[c] = __builtin_amdgcn_wmma_f32_16x16x4_f32(
          false, a, false, b, (short)0, acc[c], false, false);
    }
  }

  const int mb = (lane >> 4) << 3;  // 0 or 8
#pragma unroll
  for (int c = 0; c < 4; ++c) {
    const int col = c * 16 + m;
    const float bv = bias[col];
#pragma unroll
    for (int v = 0; v < 8; ++v)
      Y[(size_t)(row0 + mb + v) * 64 + col] = acc[c][v] + bv;
  }
}

// ---------------- edge pass 1: attention logits + segment max -----------------
// One wave per edge; lane handles features [2*lane, 2*lane+1].
__global__ __launch_bounds__(256) void edge_alpha(
    const int* __restrict__ src, const int* __restrict__ dst,
    const float* __restrict__ edge_attr, const float* __restrict__ ea_sum,
    const float* __restrict__ xl, const float* __restrict__ xr,
    const float* __restrict__ We, const float* __restrict__ att,
    float* __restrict__ alpha, unsigned* __restrict__ amax_enc) {
  const int lane = threadIdx.x & 31;
  const int e = blockIdx.x * 8 + (threadIdx.x >> 5);
  if (e >= EE + NN) return;
  int s, d;
  float ea;
  if (e < EE) {
    s = src[e];
    d = dst[e];
    ea = edge_attr[e];
  } else {
    s = d = e - EE;
    ea = ea_sum[0] * (1.0f / (float)EE);
  }
  const int f = lane << 1;
  float2 a  = *(const float2*)(xl + (size_t)s * 64 + f);
  float2 b  = *(const float2*)(xr + (size_t)d * 64 + f);
  float2 w  = *(const float2*)(We + f);
  float2 at = *(const float2*)(att + f);
  float mx = a.x + b.x + ea * w.x;
  float my = a.y + b.y + ea * w.y;
  mx = mx > 0.0f ? mx : mx * NEG_SLOPE;
  my = my > 0.0f ? my : my * NEG_SLOPE;
  float p = mx * at.x + my * at.y;
  for (int o = 16; o > 0; o >>= 1) p += __shfl_xor(p, o, 32);
  if (lane == 0) {
    alpha[e] = p;
    atomicMax(&amax_enc[d], f2ord(p));
  }
}

// ---------------- edge pass 2: exp weights + fused weighted scatter -----------
__global__ __launch_bounds__(256) void edge_accum(
    const int* __restrict__ src, const int* __restrict__ dst,
    const float* __restrict__ xl, const float* __restrict__ alpha,
    const unsigned* __restrict__ amax_enc, float* __restrict__ denom,
    float* __restrict__ acc) {
  const int lane = threadIdx.x & 31;
  const int e = blockIdx.x * 8 + (threadIdx.x >> 5);
  if (e >= EE + NN) return;
  int s, d;
  if (e < EE) {
    s = src[e];
    d = dst[e];
  } else {
    s = d = e - EE;
  }
  const float ex = expf(alpha[e] - ord2f(amax_enc[d]));
  if (lane == 0) atomicAdd(&denom[d], ex);
  const int f = lane << 1;
  float2 a = *(const float2*)(xl + (size_t)s * 64 + f);
  float* op = acc + (size_t)d * 64 + f;
  atomicAdd(op + 0, ex * a.x);
  atomicAdd(op + 1, ex * a.y);
}

// ---------------- node finalize: h = relu(acc/denom + bias) -------------------
__global__ __launch_bounds__(256) void node_finalize(
    const float* __restrict__ acc, const float* __restrict__ denom,
    const float* __restrict__ bias, float* __restrict__ h) {
  long i = (long)blockIdx.x * blockDim.x + threadIdx.x;
  if (i >= (long)NN * 64) return;
  float v = acc[i] / denom[i >> 6] + bias[i & 63];
  h[i] = v > 0.0f ? v : 0.0f;
}

// ---------------- mean pooling: column sums ----------------------------------
__global__ __launch_bounds__(256) void colsum(const float* __restrict__ h,
                                              float* __restrict__ g) {
  __shared__ float sm[64];
  if (threadIdx.x < 64) sm[threadIdx.x] = 0.0f;
  __syncthreads();
  const int f = threadIdx.x & 63;
  const int rsub = threadIdx.x >> 6;
  float a = 0.0f;
  for (long r = (long)blockIdx.x * 4 + rsub; r < NN; r += (long)gridDim.x * 4)
    a += h[r * 64 + f];
  atomicAdd(&sm[f], a);
  __syncthreads();
  if (threadIdx.x < 64) atomicAdd(&g[threadIdx.x], sm[threadIdx.x]);
}

// ---------------- head: mean -> linear -> softmax ----------------------------
__global__ void head(const float* __restrict__ g, const float* __restrict__ W_lin,
                     const float* __restrict__ b_lin, float* __restrict__ out) {
  if (threadIdx.x != 0 || blockIdx.x != 0) return;
  float l0 = b_lin[0], l1 = b_lin[1];
  for (int f = 0; f < 64; ++f) {
    float mv = g[f] * (1.0f / (float)NN);
    l0 += mv * W_lin[f * 2 + 0];
    l1 += mv * W_lin[f * 2 + 1];
  }
  float mx = fmaxf(l0, l1);
  float e0 = expf(l0 - mx), e1 = expf(l1 - mx);
  float s = e0 + e1;
  out[0] = e0 / s;
  out[1] = e1 / s;
}

extern "C" void kernel_launch(void* const* d_in, const int* in_sizes, int n_in,
                              void* d_out, int out_size, void* d_ws, size_t ws_size,
                              hipStream_t stream) {
  (void)in_sizes; (void)n_in; (void)out_size; (void)ws_size;
  const float* x         = (const float*)d_in[0];
  const int*   src       = (const int*)d_in[1];
  const int*   dst       = (const int*)d_in[1] + EE;
  const float* edge_attr = (const float*)d_in[2];
  const float* W_lin     = (const float*)d_in[24];
  const float* b_lin     = (const float*)d_in[25];
  float* out = (float*)d_out;

  // workspace layout (floats), 64-element aligned slices
  float* ws = (float*)d_ws;
  size_t off = 0;
  auto take = [&](size_t n) { float* p = ws + off; off += (n + 63) & ~(size_t)63; return p; };
  float*    xl     = take((size_t)NN * 64);
  float*    xr     = take((size_t)NN * 64);
  float*    acc    = take((size_t)NN * 64);
  float*    hbuf   = take((size_t)NN * 64);
  float*    alpha  = take((size_t)(EE + NN));
  float*    denom  = take(NN);
  unsigned* amax   = (unsigned*)take(NN);
  float*    g      = take(64);
  float*    ea_sum = take(64);

  const int ET = EE + NN;
  const dim3 eb(256);
  const dim3 eg((ET + 7) / 8);
  const dim3 gemm_b(128);
  const dim3 gemm_g((NN / 16 + 3) / 4);

  // edge_attr mean (self-loop fill)
  zero_f32<<<1, 64, 0, stream>>>(ea_sum, 1);
  zero_f32<<<1, 64, 0, stream>>>(g, 64);
  reduce_sum<<<512, 256, 0, stream>>>(edge_attr, ea_sum, EE);

  for (int layer = 0; layer < 3; ++layer) {
    const float* Wl   = (const float*)d_in[3 + 7 * layer + 0];
    const float* bl   = (const float*)d_in[3 + 7 * layer + 1];
    const float* Wr   = (const float*)d_in[3 + 7 * layer + 2];
    const float* br   = (const float*)d_in[3 + 7 * layer + 3];
    const float* We   = (const float*)d_in[3 + 7 * layer + 4];
    const float* att  = (const float*)d_in[3 + 7 * layer + 5];
    const float* bias = (const float*)d_in[3 + 7 * layer + 6];
    const float* xin = (layer == 0) ? x : hbuf;

    if (layer == 0) {
      gemm_wmma_f32<128><<<gemm_g, gemm_b, 0, stream>>>(xin, Wl, bl, xl, NN);
      gemm_wmma_f32<128><<<gemm_g, gemm_b, 0, stream>>>(xin, Wr, br, xr, NN);
    } else {
      gemm_wmma_f32<64><<<gemm_g, gemm_b, 0, stream>>>(xin, Wl, bl, xl, NN);
      gemm_wmma_f32<64><<<gemm_g, gemm_b, 0, stream>>>(xin, Wr, br, xr, NN);
    }

    zero_f32<<<4096, 256, 0, stream>>>(acc, (long)NN * 64);
    zero_f32<<<512, 256, 0, stream>>>(denom, NN);
    zero_f32<<<512, 256, 0, stream>>>((float*)amax, NN);  // 0 == ordered -inf floor

    edge_alpha<<<eg, eb, 0, stream>>>(src, dst, edge_attr, ea_sum, xl, xr, We,
                                      att, alpha, amax);
    edge_accum<<<eg, eb, 0, stream>>>(src, dst, xl, alpha, amax, denom, acc);
    node_finalize<<<((long)NN * 64 + 255) / 256, 256, 0, stream>>>(acc, denom,
                                                                   bias, hbuf);
  }

  colsum<<<1024, 256, 0, stream>>>(hbuf, g);
  head<<<1, 64, 0, stream>>>(g, W_lin, b_lin, out);
}